// QRDecomp_28887950032904
// MI455X (gfx1250) — compile-verified
//
#include <hip/hip_runtime.h>
#include <cmath>

typedef __attribute__((ext_vector_type(2))) float v2f;
typedef __attribute__((ext_vector_type(8))) float v8f;

// D = A(16x4) * B(4x16) + C, f32 WMMA. f32 WMMA has no A/B NEG per ISA
// (NEG = {CNeg,0,0}), so operand negation is done in VALU before the call.
static __device__ __forceinline__ v8f wmma4(v2f a, v2f b, v8f c) {
  return __builtin_amdgcn_wmma_f32_16x16x4_f32(false, a, false, b, (short)0, c,
                                               false, false);
}

// Generic pointer -> LDS byte offset (addrspace(3) ptrtoint).
static __device__ __forceinline__ uint32_t lds_off(const void* p) {
  return (uint32_t)(uintptr_t)(__attribute__((address_space(3))) const void*)p;
}

// One wave-level async DMA: 32 lanes x 16B = 512B global -> LDS, ASYNCcnt++.
static __device__ __forceinline__ void async_copy_b128(uint32_t lds_byte_addr,
                                                       const void* gaddr) {
  asm volatile("global_load_async_to_lds_b128 %0, %1, off"
               :: "v"(lds_byte_addr), "v"(gaddr)
               : "memory");
}

static __device__ __forceinline__ void wait_async_le2() {
  asm volatile("s_wait_asynccnt 0x2" ::: "memory");
}
static __device__ __forceinline__ void wait_async_0() {
  asm volatile("s_wait_asynccnt 0x0" ::: "memory");
}

constexpr int NB = 128;   // batches
constexpr int MM = 2048;  // rows per matrix
constexpr int KK = 128;   // cols per matrix
constexpr int GP = 129;   // LDS pitch for 128x128 matrices (pad 1)
constexpr int SP = 132;   // LDS pitch for row staging (pad 4)

constexpr int MAT = 128 * GP;        // 16512 floats per component
constexpr int PP_OFF = 2 * MAT;      // 33024: ping-pong / phase-4 staging
constexpr int CH_SZ = 8 * SP;        // 1056 floats: 8 rows, one component
constexpr int BUF_SZ = 2 * CH_SZ;    // 2112 floats: one ping-pong buffer
constexpr int INVD_OFF = PP_OFF + 2 * BUF_SZ;  // 37248
constexpr int SMEM_FLOATS = INVD_OFF + 128;    // 37376 floats ~ 146 KB

__global__ __launch_bounds__(256, 1)
void cholqr_c64_kernel(const float* __restrict__ gwre,
                       const float* __restrict__ gwim,
                       float2* __restrict__ gq) {
  __shared__ float smem[SMEM_FLOATS];

  const int b    = blockIdx.x;
  const int tid  = threadIdx.x;
  const int wave = tid >> 5;
  const int lane = tid & 31;
  const int lh   = lane & 15;   // lane within half-wave
  const int hs   = lane >> 4;   // half-wave select (0/1)

  const float* wre = gwre + (size_t)b * MM * KK;
  const float* wim = gwim + (size_t)b * MM * KK;

  float* const sGre = smem;            // 128x128 (pitch GP): G -> L -> U
  float* const sGim = smem + MAT;
  float* const pp   = smem + PP_OFF;   // 2 x (8 rows re + 8 rows im)
  float* const invd = smem + INVD_OFF; // 1/L[j][j] (real)

  const v8f vz = {0.f, 0.f, 0.f, 0.f, 0.f, 0.f, 0.f, 0.f};

  // ---------------- Phase 1: Gram  G = W^H W (complex) ----------------
  // 8-row chunks, double-buffered with async global->LDS DMA.
  // Per chunk each wave issues 2 async b128 ops (row `wave`, re + im).
  v8f cre[8], cim[8];
#pragma unroll
  for (int j = 0; j < 8; ++j) { cre[j] = vz; cim[j] = vz; }

  constexpr int NCH = MM / 8;  // 256 chunks
  {
    // prologue: stage chunk 0 into buffer 0
    float* bRe = pp;
    float* bIm = pp + CH_SZ;
    async_copy_b128(lds_off(bRe + wave * SP) + lane * 16,
                    wre + (size_t)wave * KK + lane * 4);
    async_copy_b128(lds_off(bIm + wave * SP) + lane * 16,
                    wim + (size_t)wave * KK + lane * 4);
  }
  for (int c = 0; c < NCH; ++c) {
    if (c + 1 < NCH) {
      const int m1 = (c + 1) * 8;
      float* nRe = pp + ((c + 1) & 1) * BUF_SZ;
      float* nIm = nRe + CH_SZ;
      async_copy_b128(lds_off(nRe + wave * SP) + lane * 16,
                      wre + (size_t)(m1 + wave) * KK + lane * 4);
      async_copy_b128(lds_off(nIm + wave * SP) + lane * 16,
                      wim + (size_t)(m1 + wave) * KK + lane * 4);
      wait_async_le2();  // chunk c's 2 ops done; 2 newer outstanding
    } else {
      wait_async_0();
    }
    __syncthreads();  // chunk c visible to all waves

    const float* bRe = pp + (c & 1) * BUF_SZ;
    const float* bIm = bRe + CH_SZ;
#pragma unroll
    for (int q = 0; q < 8; q += 4) {
      // A fragment = (W block)^T: lanes0-15 K={0,1}, lanes16-31 K={2,3}
      const int ar = q + (hs ? 2 : 0);
      const int ac = wave * 16 + lh;
      v2f Are = {bRe[ar * SP + ac], bRe[(ar + 1) * SP + ac]};
      v2f Aim = {bIm[ar * SP + ac], bIm[(ar + 1) * SP + ac]};
      v2f nAim = -Aim;  // conj(W^H) via sign on the imag products
#pragma unroll
      for (int j = 0; j < 8; ++j) {
        const int bc = j * 16 + lh;
        v2f Bre = {bRe[ar * SP + bc], bRe[(ar + 1) * SP + bc]};
        v2f Bim = {bIm[ar * SP + bc], bIm[(ar + 1) * SP + bc]};
        // Gre += Wre^T Wre + Wim^T Wim ; Gim += Wre^T Wim - Wim^T Wre
        cre[j] = wmma4(Are, Bre, cre[j]);
        cre[j] = wmma4(Aim, Bim, cre[j]);
        cim[j] = wmma4(Are, Bim, cim[j]);
        cim[j] = wmma4(nAim, Bre, cim[j]);
      }
    }
    __syncthreads();  // all reads done before buffer is overwritten
  }
  // C layout: VGPR r -> row r (lanes0-15) / row r+8 (lanes16-31), col = lane%16
#pragma unroll
  for (int j = 0; j < 8; ++j) {
#pragma unroll
    for (int r = 0; r < 8; ++r) {
      const int row = wave * 16 + r + hs * 8;
      const int col = j * 16 + lh;
      sGre[row * GP + col] = cre[j][r];
      sGim[row * GP + col] = cim[j][r];
    }
  }
  __syncthreads();

  // ---------------- Phase 2: Cholesky G = L L^H (in place, lower) -----
  for (int j = 0; j < KK; ++j) {
    if (tid < 32) {
      float s = 0.f;
      for (int k = lane; k < j; k += 32) {
        const float a = sGre[j * GP + k], c = sGim[j * GP + k];
        s += a * a + c * c;
      }
#pragma unroll
      for (int off = 16; off > 0; off >>= 1) s += __shfl_down(s, off, 32);
      if (lane == 0) {
        float d = sGre[j * GP + j] - s;
        d = d > 1e-30f ? d : 1e-30f;
        const float l = sqrtf(d);
        sGre[j * GP + j] = l;
        sGim[j * GP + j] = 0.f;
        invd[j] = 1.0f / l;
      }
    }
    __syncthreads();
    const int i = j + 1 + tid;
    if (i < KK) {
      float accr = sGre[i * GP + j], acci = sGim[i * GP + j];
      for (int k = 0; k < j; ++k) {
        const float lr = sGre[i * GP + k], li = sGim[i * GP + k];
        const float mr = sGre[j * GP + k], mi = sGim[j * GP + k];
        accr -= lr * mr + li * mi;  // L[i][k] * conj(L[j][k])
        acci -= li * mr - lr * mi;
      }
      const float inv = invd[j];
      sGre[i * GP + j] = accr * inv;
      sGim[i * GP + j] = acci * inv;
    }
    __syncthreads();
  }

  // ---------------- Phase 3: U = L^-H into strictly-upper sG ----------
  // Thread j computes column j of Linv; U[j][i] = conj(Linv[i][j]).
  if (tid < KK) {
    const int j = tid;
    const float xjr = invd[j];  // Linv[j][j] (real), kept in a register
    for (int i = j + 1; i < KK; ++i) {
      float lr = sGre[i * GP + j], li = sGim[i * GP + j];
      float sr = lr * xjr, si = li * xjr;  // k == j term
      for (int k = j + 1; k < i; ++k) {
        lr = sGre[i * GP + k];
        li = sGim[i * GP + k];
        const float xr = sGre[j * GP + k];   // x[k] = conj(U[j][k])
        const float xi = -sGim[j * GP + k];
        sr += lr * xr - li * xi;
        si += lr * xi + li * xr;
      }
      const float inv = invd[i];
      sGre[j * GP + i] = -sr * inv;  // U[j][i] = conj(-s / L[i][i])
      sGim[j * GP + i] = si * inv;
    }
  }
  __syncthreads();
  // Turn sG into dense U: zero strictly-lower, diag = 1/L[j][j].
  for (int i = tid; i < 128 * 128; i += 256) {
    const int r = i >> 7, c = i & 127;
    if (r > c) {
      sGre[r * GP + c] = 0.f;
      sGim[r * GP + c] = 0.f;
    }
  }
  if (tid < KK) {
    sGre[tid * GP + tid] = invd[tid];
    sGim[tid * GP + tid] = 0.f;
  }
  __syncthreads();

  // ---------------- Phase 4: Q = W * U (complex GEMM), store ----------
  // 16-row blocks; async-stage 32 rows (re+im), 4 DMA ops per wave.
  float* const gsRe = pp;           // 16 x SP
  float* const gsIm = pp + 16 * SP; // 16 x SP
  for (int rb = 0; rb < 128; ++rb) {
    const int row0 = rb * 16;
    __syncthreads();  // prior reads of staging done
    {
      const int r1 = wave, r2 = wave + 8;
      async_copy_b128(lds_off(gsRe + r1 * SP) + lane * 16,
                      wre + (size_t)(row0 + r1) * KK + lane * 4);
      async_copy_b128(lds_off(gsRe + r2 * SP) + lane * 16,
                      wre + (size_t)(row0 + r2) * KK + lane * 4);
      async_copy_b128(lds_off(gsIm + r1 * SP) + lane * 16,
                      wim + (size_t)(row0 + r1) * KK + lane * 4);
      async_copy_b128(lds_off(gsIm + r2 * SP) + lane * 16,
                      wim + (size_t)(row0 + r2) * KK + lane * 4);
    }
    wait_async_0();
    __syncthreads();

    v8f qre = vz, qim = vz;
    const int bc = wave * 16 + lh;  // wave owns column tile `wave`
#pragma unroll 4
    for (int k0 = 0; k0 < KK; k0 += 4) {
      const int acol = k0 + (hs ? 2 : 0);
      v2f Are = {gsRe[lh * SP + acol], gsRe[lh * SP + acol + 1]};
      v2f Aim = {gsIm[lh * SP + acol], gsIm[lh * SP + acol + 1]};
      v2f nAim = -Aim;
      const int brow = acol;
      v2f Bre = {sGre[brow * GP + bc], sGre[(brow + 1) * GP + bc]};
      v2f Bim = {sGim[brow * GP + bc], sGim[(brow + 1) * GP + bc]};
      // Qre += Wre*Ure - Wim*Uim ; Qim += Wre*Uim + Wim*Ure
      qre = wmma4(Are, Bre, qre);
      qre = wmma4(nAim, Bim, qre);
      qim = wmma4(Are, Bim, qim);
      qim = wmma4(Aim, Bre, qim);
    }
#pragma unroll
    for (int r = 0; r < 8; ++r) {
      const int row = row0 + r + hs * 8;
      const int col = wave * 16 + lh;
      gq[((size_t)b * MM + row) * KK + col] = make_float2(qre[r], qim[r]);
    }
  }
}

extern "C" void kernel_launch(void* const* d_in, const int* in_sizes, int n_in,
                              void* d_out, int out_size, void* d_ws,
                              size_t ws_size, hipStream_t stream) {
  (void)in_sizes; (void)n_in; (void)d_ws; (void)ws_size; (void)out_size;
  const float* w_re = (const float*)d_in[0];
  const float* w_im = (const float*)d_in[1];
  float2* q = (float2*)d_out;  // complex64 -> interleaved (re, im) floats
  cholqr_c64_kernel<<<dim3(NB), dim3(256), 0, stream>>>(w_re, w_im, q);
}